// DSVABlock_52785148068469
// MI455X (gfx1250) — compile-verified
//
#include <hip/hip_runtime.h>
#include <hip/hip_bf16.h>
#include <math.h>

// ---------------- problem constants (match reference setup) ----------------
#define NB    4
#define NPTS  4096
#define CDIM  64
#define MDIM  256
#define HNUM  4
#define DH    16
#define KNN_  10
#define KS_   4
#define ROWS  (NB * NPTS)          // 16384 flat tokens

typedef __attribute__((ext_vector_type(2))) float v2f;
typedef __attribute__((ext_vector_type(8))) float v8f;

__device__ __forceinline__ float gelu_tanh(float v) {
  // jax.nn.gelu default (approximate=True)
  const float c = 0.7978845608028654f;
  float u = c * (v + 0.044715f * v * v * v);
  return 0.5f * v * (1.0f + tanhf(u));
}

// ---------------- LayerNorm over C=64, one wave (32 lanes) per row ----------
__global__ void ln64_kernel(const float* __restrict__ x, const float* __restrict__ g,
                            const float* __restrict__ bta, float* __restrict__ y, int rows) {
  int wave = blockIdx.x * (blockDim.x >> 5) + (threadIdx.x >> 5);
  int lane = threadIdx.x & 31;
  if (wave >= rows) return;
  const float* xr = x + (size_t)wave * CDIM;
  float v0 = xr[lane], v1 = xr[lane + 32];
  float s = v0 + v1;
#pragma unroll
  for (int off = 16; off > 0; off >>= 1) s += __shfl_xor(s, off, 32);
  float mean = s * (1.0f / 64.0f);
  float d0 = v0 - mean, d1 = v1 - mean;
  float q = d0 * d0 + d1 * d1;
#pragma unroll
  for (int off = 16; off > 0; off >>= 1) q += __shfl_xor(q, off, 32);
  float inv = rsqrtf(q * (1.0f / 64.0f) + 1e-5f);
  float* yr = y + (size_t)wave * CDIM;
  yr[lane]      = d0 * inv * g[lane]      + bta[lane];
  yr[lane + 32] = d1 * inv * g[lane + 32] + bta[lane + 32];
}

// ---------------- f32 WMMA GEMM: out[rows x Nout] = A[rows x K] @ W[K x Nout] + bias
// One wave computes one 16x16 tile via V_WMMA_F32_16X16X4_F32 (K stepped by 4).
// A-frag (16x4): lane l<16 holds A[m=l][2g..2g+1] in v[0..1], g = lane>>4.
// B-frag (4x16): lane holds B[2g][n], B[2g+1][n], n = lane&15.
// C/D (16x16):  acc[r] = D[r + 8g][n].
// MODE 0: plain (+bias)                       -> qkv
// MODE 1: out = inputs + (mask ? 0.5*v : 0)   -> x2  (proj + residual)
// MODE 2: out = gelu(v)                       -> h1
// MODE 3: out = 0.5*v + res                   -> final output
template <int MODE>
__global__ void wmma_gemm_kernel(const float* __restrict__ A, const float* __restrict__ W,
                                 const float* __restrict__ bias, float* __restrict__ out,
                                 int K, int Nout, int tilesN, int tilesTotal,
                                 const float* __restrict__ res,
                                 const unsigned char* __restrict__ rmask) {
  int wave = blockIdx.x * (blockDim.x >> 5) + (threadIdx.x >> 5);
  if (wave >= tilesTotal) return;
  int tm = wave / tilesN, tn = wave % tilesN;
  int lane = threadIdx.x & 31;
  int g = lane >> 4;          // half-wave selector (K pair)
  int l = lane & 15;
  int arow = tm * 16 + l;     // A row for this lane's fragment
  int bcol = tn * 16 + l;     // B/out column for this lane
  const float* Ar = A + (size_t)arow * K;

  v8f acc = {};
  for (int kk = 0; kk < K; kk += 4) {
    v2f a = *(const v2f*)&Ar[kk + 2 * g];                 // 8B-aligned (kk+2g even)
    v2f b;
    b.x = W[(size_t)(kk + 2 * g) * Nout + bcol];
    b.y = W[(size_t)(kk + 2 * g + 1) * Nout + bcol];
    acc = __builtin_amdgcn_wmma_f32_16x16x4_f32(false, a, false, b, (short)0, acc,
                                                false, false);
  }

  float bv = bias[bcol];
#pragma unroll
  for (int r = 0; r < 8; ++r) {
    int row = tm * 16 + r + 8 * g;
    float v = acc[r] + bv;
    size_t oidx = (size_t)row * Nout + bcol;
    if (MODE == 0) {
      out[oidx] = v;
    } else if (MODE == 1) {
      float a0 = rmask[row] ? 0.5f * v : 0.0f;  // where(mask, proj, 0), then 0.5*a + x
      out[oidx] = a0 + res[oidx];
    } else if (MODE == 2) {
      out[oidx] = gelu_tanh(v);
    } else {
      out[oidx] = 0.5f * v + res[oidx];
    }
  }
}

// ---------------- KNN: top-10 nearest by squared distance, masked -> +1e9 ----
// 256 queries per block; the batch's 4096 coords + mask penalty live in LDS (64KB).
__global__ void knn_kernel(const float* __restrict__ coords,
                           const unsigned char* __restrict__ mask,
                           int* __restrict__ nbr) {
  __shared__ float sx[NPTS], sy[NPTS], sz[NPTS], sm[NPTS];
  int b = blockIdx.x >> 4;            // 16 blocks per batch (4096/256)
  int qblk = blockIdx.x & 15;
  const float* cb = coords + (size_t)b * NPTS * 3;
  const unsigned char* mb = mask + (size_t)b * NPTS;
  for (int j = threadIdx.x; j < NPTS; j += blockDim.x) {
    sx[j] = cb[j * 3 + 0];
    sy[j] = cb[j * 3 + 1];
    sz[j] = cb[j * 3 + 2];
    sm[j] = mb[j] ? 0.0f : 1e9f;      // masked voxels pushed past any real d2 (max 768)
  }
  __syncthreads();

  int qi = qblk * 256 + threadIdx.x;
  float qx = sx[qi], qy = sy[qi], qz = sz[qi];
  float d[KNN_];
  int id[KNN_];
#pragma unroll
  for (int t = 0; t < KNN_; ++t) { d[t] = 3.4e38f; id[t] = 0; }

  for (int j = 0; j < NPTS; ++j) {
    float dx = sx[j] - qx, dy = sy[j] - qy, dz = sz[j] - qz;
    float d2 = dx * dx + dy * dy + dz * dz + sm[j];
    if (d2 < d[KNN_ - 1]) {           // strict < : ties keep lower index (JAX top_k)
#pragma unroll
      for (int p = KNN_ - 1; p >= 0; --p) {
        bool shift = (p > 0) && (d2 < d[p - 1]);
        bool place = (d2 < d[p]) && !shift;
        float nd = shift ? d[p - 1] : (place ? d2 : d[p]);
        int ni = shift ? id[p - 1] : (place ? j : id[p]);
        d[p] = nd;
        id[p] = ni;
      }
    }
  }
  int* o = nbr + ((size_t)b * NPTS + qi) * KNN_;
#pragma unroll
  for (int t = 0; t < KNN_; ++t) o[t] = id[t];
}

// ---------------- attention: thread per (b, n, h) --------------------------
// qkv layout [B*N, 192]: q at h*16, k at 64+h*16, v at 128+h*16.
__global__ void attn_kernel(const float* __restrict__ qkv, const int* __restrict__ nbr,
                            const unsigned char* __restrict__ mask,
                            float* __restrict__ o) {
  int t = blockIdx.x * blockDim.x + threadIdx.x;
  if (t >= ROWS * HNUM) return;
  int h = t & (HNUM - 1);
  int bn = t >> 2;                    // flat b*N + n
  int b = bn >> 12;                   // N = 4096

  const float* qrow = qkv + (size_t)bn * (3 * CDIM) + h * DH;
  float q[DH];
#pragma unroll
  for (int dd = 0; dd < DH; ++dd) q[dd] = qrow[dd];

  const int* nb = nbr + (size_t)bn * KNN_;
  int nj[KNN_];
  float sc[KNN_];
#pragma unroll
  for (int k = 0; k < KNN_; ++k) {
    nj[k] = nb[k];
    const float* krow = qkv + ((size_t)b * NPTS + nj[k]) * (3 * CDIM) + CDIM + h * DH;
    float s = 0.0f;
#pragma unroll
    for (int dd = 0; dd < DH; ++dd) s += q[dd] * krow[dd];
    s *= 0.25f;                       // 1/sqrt(16)
    if (!mask[(size_t)b * NPTS + nj[k]]) s -= 1e9f;
    sc[k] = s;
  }

  // top-4 by score, descending; strict > keeps earlier (lower) index on ties
  float sv[KS_];
  int si[KS_];
#pragma unroll
  for (int p = 0; p < KS_; ++p) { sv[p] = -3.4e38f; si[p] = 0; }
#pragma unroll
  for (int k = 0; k < KNN_; ++k) {
    float s = sc[k];
    if (s > sv[KS_ - 1]) {
#pragma unroll
      for (int p = KS_ - 1; p >= 0; --p) {
        bool shift = (p > 0) && (s > sv[p - 1]);
        bool place = (s > sv[p]) && !shift;
        float nv = shift ? sv[p - 1] : (place ? s : sv[p]);
        int ni = shift ? si[p - 1] : (place ? k : si[p]);
        sv[p] = nv;
        si[p] = ni;
      }
    }
  }

  float mx = sv[0];                   // sorted desc -> max
  float e[KS_], es = 0.0f;
#pragma unroll
  for (int p = 0; p < KS_; ++p) { e[p] = expf(sv[p] - mx); es += e[p]; }
  float inv = 1.0f / es;

  float acc[DH];
#pragma unroll
  for (int dd = 0; dd < DH; ++dd) acc[dd] = 0.0f;
#pragma unroll
  for (int p = 0; p < KS_; ++p) {
    const float* vrow =
        qkv + ((size_t)b * NPTS + nj[si[p]]) * (3 * CDIM) + 2 * CDIM + h * DH;
    float w = e[p] * inv;
#pragma unroll
    for (int dd = 0; dd < DH; ++dd) acc[dd] += w * vrow[dd];
  }
  float* orow = o + (size_t)bn * CDIM + h * DH;   // [B,N,H*dh] => proj-ready layout
#pragma unroll
  for (int dd = 0; dd < DH; ++dd) orow[dd] = acc[dd];
}

// ---------------- driver -----------------------------------------------------
extern "C" void kernel_launch(void* const* d_in, const int* in_sizes, int n_in,
                              void* d_out, int out_size, void* d_ws, size_t ws_size,
                              hipStream_t stream) {
  const float* x      = (const float*)d_in[0];
  const float* coords = (const float*)d_in[1];
  const unsigned char* mask = (const unsigned char*)d_in[2];   // jnp bool_ = 1 byte
  const float* ln1_g  = (const float*)d_in[3];
  const float* ln1_b  = (const float*)d_in[4];
  const float* ln2_g  = (const float*)d_in[5];
  const float* ln2_b  = (const float*)d_in[6];
  const float* w_qkv  = (const float*)d_in[7];
  const float* b_qkv  = (const float*)d_in[8];
  const float* w_proj = (const float*)d_in[9];
  const float* b_proj = (const float*)d_in[10];
  const float* w1     = (const float*)d_in[11];
  const float* b1     = (const float*)d_in[12];
  const float* w2     = (const float*)d_in[13];
  const float* b2     = (const float*)d_in[14];
  float* outp = (float*)d_out;

  // workspace carve-up (~47 MB of f32)
  float* ws   = (float*)d_ws;
  float* xln  = ws;                                   // [ROWS, 64]
  float* qkv  = xln + (size_t)ROWS * CDIM;            // [ROWS, 192]
  int*   nbr  = (int*)(qkv + (size_t)ROWS * 3 * CDIM);// [ROWS, 10]
  float* oatt = (float*)(nbr + (size_t)ROWS * KNN_);  // [ROWS, 64]
  float* x2   = oatt + (size_t)ROWS * CDIM;           // [ROWS, 64]
  float* xln2 = x2 + (size_t)ROWS * CDIM;             // [ROWS, 64]
  float* h1   = xln2 + (size_t)ROWS * CDIM;           // [ROWS, 256]

  // 1) LN1
  ln64_kernel<<<ROWS / 4, 128, 0, stream>>>(x, ln1_g, ln1_b, xln, ROWS);

  // 2) QKV projection: [16384,64] @ [64,192]
  {
    int tilesN = (3 * CDIM) / 16, tiles = (ROWS / 16) * tilesN;
    wmma_gemm_kernel<0><<<(tiles + 3) / 4, 128, 0, stream>>>(
        xln, w_qkv, b_qkv, qkv, CDIM, 3 * CDIM, tilesN, tiles, nullptr, nullptr);
  }

  // 3) KNN top-10 over coords (LDS-staged brute force, 4096^2 per batch)
  knn_kernel<<<NB * (NPTS / 256), 256, 0, stream>>>(coords, mask, nbr);

  // 4) sparse dynamic attention (top-4 of 10, softmax, gather-v)
  {
    int tot = ROWS * HNUM;
    attn_kernel<<<(tot + 255) / 256, 256, 0, stream>>>(qkv, nbr, mask, oatt);
  }

  // 5) output projection + mask + residual: x2 = x + (mask ? 0.5*(o@Wp+bp) : 0)
  {
    int tilesN = CDIM / 16, tiles = (ROWS / 16) * tilesN;
    wmma_gemm_kernel<1><<<(tiles + 3) / 4, 128, 0, stream>>>(
        oatt, w_proj, b_proj, x2, CDIM, CDIM, tilesN, tiles, x, mask);
  }

  // 6) LN2
  ln64_kernel<<<ROWS / 4, 128, 0, stream>>>(x2, ln2_g, ln2_b, xln2, ROWS);

  // 7) MLP fc1 + GELU: [16384,64] @ [64,256]
  {
    int tilesN = MDIM / 16, tiles = (ROWS / 16) * tilesN;
    wmma_gemm_kernel<2><<<(tiles + 3) / 4, 128, 0, stream>>>(
        xln2, w1, b1, h1, CDIM, MDIM, tilesN, tiles, nullptr, nullptr);
  }

  // 8) MLP fc2 + residual: out = 0.5*(h1@w2+b2) + x2
  {
    int tilesN = CDIM / 16, tiles = (ROWS / 16) * tilesN;
    wmma_gemm_kernel<3><<<(tiles + 3) / 4, 128, 0, stream>>>(
        h1, w2, b2, outp, MDIM, CDIM, tilesN, tiles, x2, nullptr);
  }
}